// Transformer_84464826843547
// MI455X (gfx1250) — compile-verified
//
#include <hip/hip_runtime.h>
#include <cstdint>

// CDNA5 / gfx1250: wave32, WMMA f32 16x16x4 (exact fp32 matrix path),
// async Global->LDS DMA (ASYNCcnt) with double-buffered key-block staging.
typedef __attribute__((ext_vector_type(2))) float v2f;
typedef __attribute__((ext_vector_type(8))) float v8f;

#define BB    64
#define NP1   1025
#define DD    256
#define DP1   257
#define NLAY  4
#define NHEAD 4
#define PADN  1088   // tokens padded: 17 blocks of 64 (68 tiles of 16)
#define PADD  288    // features padded: 18 tiles of 16
#define LDSW  292    // LDS row stride (floats): 73*float4, conflict-free strided reads

static __device__ inline v8f wmma_f32_4(v2f a, v2f b, v8f c) {
  // D = A(16x4,f32) * B(4x16,f32) + C(16x16,f32)
  return __builtin_amdgcn_wmma_f32_16x16x4_f32(false, a, false, b, (short)0, c, false, false);
}

static __device__ inline v8f v8zero() {
  v8f z;
#pragma unroll
  for (int i = 0; i < 8; ++i) z[i] = 0.0f;
  return z;
}

// Async DMA of one 64x288 f32 block (row-major, stride PADD) into LDS rows of
// stride LDSW, 16B per lane per op, 18 ops per wave. Tracked by ASYNCcnt.
// gbase must be uniform (SGPR pair); per-lane 32-bit byte offset in VGPR;
// LDS byte address = low 32 bits of the generic pointer (aperture rule).
static __device__ inline void stage_async_64x288(const float* gbase, uint32_t ldsBase, int tid) {
#pragma unroll
  for (int it = 0; it < 18; ++it) {
    int t = tid + it * 256;
    int r = t / (PADD / 4);
    int q = t - r * (PADD / 4);
    uint32_t voff = (uint32_t)(r * (PADD * 4) + q * 16);
    uint32_t doff = ldsBase + (uint32_t)(r * (LDSW * 4) + q * 16);
    asm volatile("global_load_async_to_lds_b128 %0, %1, %2"
                 :: "v"(doff), "v"(voff), "s"(gbase)
                 : "memory");
  }
}

// ---------------------------------------------------------------- pad Z -> Xp
__global__ __launch_bounds__(256) void pad_z_kernel(const float* __restrict__ Z,
                                                    float* __restrict__ Xp) {
  size_t idx = (size_t)blockIdx.x * 256 + threadIdx.x;   // grid covers BB*PADN*PADD exactly
  int i = (int)(idx % PADD);
  size_t t = idx / PADD;
  int n = (int)(t % PADN);
  int b = (int)(t / PADN);
  float v = 0.0f;
  if (n < NP1 && i < DP1) v = Z[((size_t)b * NP1 + n) * DP1 + i];
  Xp[idx] = v;
}

// ---------------------------------------------- build padded Qf and Pf^T (all l,h)
__global__ __launch_bounds__(256) void pad_pq_kernel(const float* __restrict__ ap,
                                                     float* __restrict__ Qp,
                                                     float* __restrict__ Pt) {
  size_t idx = (size_t)blockIdx.x * 256 + threadIdx.x;   // grid covers 16*PADD*PADD exactly
  int j = (int)(idx % PADD);
  size_t t = idx / PADD;
  int i = (int)(t % PADD);
  int lh = (int)(t / PADD);
  const float* a0 = ap + ((size_t)lh * 2 + 0) * DD * DD;   // P[l,h]
  const float* a1 = a0 + (size_t)DD * DD;                   // Q[l,h]
  float q = (i < DD && j < DD) ? a1[(size_t)i * DD + j] : 0.0f;
  float p = (i < DD && j < DD) ? a0[(size_t)j * DD + i]
                               : ((i == DD && j == DD) ? 1.0f : 0.0f);
  Qp[idx] = q;
  Pt[idx] = p;
}

// --------------------------------------------- Y = X * W  (per batch), optional row-1024 mask
__global__ __launch_bounds__(256) void proj_kernel(const float* __restrict__ X,
                                                   const float* __restrict__ W,
                                                   float* __restrict__ Y,
                                                   int maskRow) {
  const int wave = threadIdx.x >> 5, lane = threadIdx.x & 31;
  const int l15 = lane & 15;
  const int half2 = (lane >> 4) << 1;         // K sub-offset: 0 (lanes 0-15) / 2 (lanes 16-31)
  const int tile = blockIdx.x * 8 + wave;     // 68*18 = 1224 tiles, 153 blocks exactly
  const int rt = tile / (PADD / 16);
  const int ct = tile - rt * (PADD / 16);

  const float* xg = X + ((size_t)blockIdx.y * PADN + rt * 16 + l15) * PADD + half2;
  const float* wg = W + (size_t)half2 * PADD + ct * 16 + l15;

  v8f c = v8zero();
#pragma unroll 4
  for (int k = 0; k < PADD; k += 4) {
    v2f a;  a.x = xg[k];                 a.y = xg[k + 1];
    v2f bb; bb.x = wg[(size_t)k * PADD]; bb.y = wg[(size_t)(k + 1) * PADD];
    c = wmma_f32_4(a, bb, c);
  }

  const int row0 = rt * 16 + ((lane >> 4) << 3);
  float* yg = Y + ((size_t)blockIdx.y * PADN + row0) * PADD + ct * 16 + l15;
#pragma unroll
  for (int r = 0; r < 8; ++r) {
    float v = c[r];
    if (maskRow && (row0 + r) == (NP1 - 1)) v = 0.0f;   // mask key token N
    yg[(size_t)r * PADD] = v;
  }
}

// ------------- res(+)= relu(ZQ * X^T) * Kb  — flash-style, 64 query rows per workgroup
// 8 waves: wave = (wn 0..3 -> 16-row tile, wi 0..1 -> 144-col half). 9 persistent C tiles/wave.
// Key blocks double-buffered via async Global->LDS DMA overlapped with WMMA.
__global__ __launch_bounds__(256) void attn_kernel(const float* __restrict__ ZQ,
                                                   const float* __restrict__ X,
                                                   const float* __restrict__ Kb,
                                                   float* __restrict__ R,
                                                   int accum) {
  extern __shared__ float smem[];
  float* s_zq = smem;                 // 64 x LDSW  (query-side ZQ block)
  float* s_x0 = smem + 64 * LDSW;     // ping
  float* s_x1 = s_x0 + 64 * LDSW;     // pong
  float* s_at = s_x1 + 64 * LDSW;     // 64 x 68    (attn tile, C-layout -> A-layout bounce)

  const int tid = threadIdx.x, wave = tid >> 5, lane = tid & 31;
  const int wn = wave & 3, wi = wave >> 2;
  const int l15 = lane & 15;
  const int half2 = (lane >> 4) << 1;
  const int half8 = (lane >> 4) << 3;
  const int b = blockIdx.y, nb = blockIdx.x;

  const float* xbase = X + (size_t)b * PADN * PADD;
  const uint32_t x0off = (uint32_t)(uintptr_t)s_x0;   // LDS byte address (low 32 bits)
  const uint32_t x1off = (uint32_t)(uintptr_t)s_x1;

  // prologue: async-stage key block 0
  stage_async_64x288(xbase, x0off, tid);

  // stage ZQ block (64 x 288) once, plain path
  const float* zg = ZQ + ((size_t)b * PADN + nb * 64) * PADD;
  for (int t = tid; t < 64 * (PADD / 4); t += 256) {
    int r = t / (PADD / 4), q = t - r * (PADD / 4);
    *(float4*)(s_zq + r * LDSW + q * 4) = *(const float4*)(zg + (size_t)r * PADD + q * 4);
  }

  v8f acc[9];
#pragma unroll
  for (int t = 0; t < 9; ++t) acc[t] = v8zero();

  for (int mb = 0; mb < PADN / 64; ++mb) {
    float* cur = (mb & 1) ? s_x1 : s_x0;
    if (mb + 1 < PADN / 64) {
      // kick off DMA for the next key block into the other buffer, then wait
      // for the current block only (async loads complete in order).
      stage_async_64x288(xbase + (size_t)(mb + 1) * 64 * PADD,
                         (mb & 1) ? x0off : x1off, tid);
      asm volatile("s_wait_asynccnt 0x12" ::: "memory");   // <=18 outstanding => block mb done
    } else {
      asm volatile("s_wait_asynccnt 0x0" ::: "memory");
    }
    __syncthreads();   // all waves' DMA (and ZQ staging / prior s_at reads) done

    // GEMM1: attn(64x64) = relu( zq(64x288) * cur(64x288)^T ); wave: n-tile wn, m-tiles {2wi, 2wi+1}
    v8f c0 = v8zero(), c1 = v8zero();
    const float* ap  = s_zq + (wn * 16 + l15) * LDSW + half2;
    const float* bp0 = cur + (wi * 32 + l15) * LDSW + half2;
    const float* bp1 = bp0 + 16 * LDSW;
#pragma unroll 4
    for (int k = 0; k < PADD; k += 4) {
      v2f a;  a.x  = ap[k];  a.y  = ap[k + 1];
      v2f b0; b0.x = bp0[k]; b0.y = bp0[k + 1];
      v2f b1; b1.x = bp1[k]; b1.y = bp1[k + 1];
      c0 = wmma_f32_4(a, b0, c0);
      c1 = wmma_f32_4(a, b1, c1);
    }
    const int arow = wn * 16 + half8;
#pragma unroll
    for (int r = 0; r < 8; ++r) {
      s_at[(arow + r) * 68 + wi * 32 + l15]      = fmaxf(c0[r], 0.0f);
      s_at[(arow + r) * 68 + wi * 32 + 16 + l15] = fmaxf(c1[r], 0.0f);
    }
    __syncthreads();   // attn tile complete before cross-wave reads (also fences buffer reuse)

    // GEMM2: acc += attn(16x64) * Kb(64x288) — wave covers cols [wi*144, wi*144+143]
    const float* ap2 = s_at + (wn * 16 + l15) * 68 + half2;
    const float* kg  = Kb + ((size_t)b * PADN + mb * 64 + half2) * PADD + wi * 144 + l15;
    for (int kk = 0; kk < 64; kk += 4) {
      v2f a; a.x = ap2[kk]; a.y = ap2[kk + 1];
      const float* kgr = kg + (size_t)kk * PADD;
#pragma unroll
      for (int t = 0; t < 9; ++t) {
        v2f bb; bb.x = kgr[t * 16]; bb.y = kgr[t * 16 + PADD];
        acc[t] = wmma_f32_4(a, bb, acc[t]);
      }
    }
  }

  // epilogue: res (head 0 overwrites, later heads accumulate; stream-ordered, no races)
  float* rg = R + ((size_t)b * PADN + nb * 64 + wn * 16 + half8) * PADD + wi * 144 + l15;
#pragma unroll
  for (int t = 0; t < 9; ++t) {
#pragma unroll
    for (int r = 0; r < 8; ++r) {
      size_t off = (size_t)r * PADD + t * 16;
      float v = acc[t][r];
      if (accum) v += rg[off];
      rg[off] = v;
    }
  }
}

// ---------------------------------------------------------------- Z += res / N
__global__ __launch_bounds__(256) void resid_kernel(float* __restrict__ Xp,
                                                    const float* __restrict__ R) {
  size_t idx = (size_t)blockIdx.x * 256 + threadIdx.x;   // exact grid
  Xp[idx] += R[idx] * (1.0f / (float)(NP1 - 1));
}

// ---------------------------------------------------------------- unpad -> out
__global__ __launch_bounds__(256) void unpad_kernel(const float* __restrict__ Xp,
                                                    float* __restrict__ out) {
  size_t idx = (size_t)blockIdx.x * 256 + threadIdx.x;
  if (idx >= (size_t)BB * NP1 * DP1) return;
  int i = (int)(idx % DP1);
  size_t t = idx / DP1;
  int n = (int)(t % NP1);
  int b = (int)(t / NP1);
  out[idx] = Xp[((size_t)b * PADN + n) * PADD + i];
}

extern "C" void kernel_launch(void* const* d_in, const int* in_sizes, int n_in,
                              void* d_out, int out_size, void* d_ws, size_t ws_size,
                              hipStream_t stream) {
  const float* Z  = (const float*)d_in[0];   // (64,1025,257) f32
  const float* ap = (const float*)d_in[1];   // (4,4,2,256,256) f32
  float* out = (float*)d_out;

  const size_t XPE = (size_t)BB * PADN * PADD;        // floats per big buffer
  const size_t QPE = (size_t)NLAY * NHEAD * PADD * PADD;
  float* Xp = (float*)d_ws;
  float* ZQ = Xp + XPE;
  float* Kb = ZQ + XPE;
  float* Rs = Kb + XPE;
  float* Qp = Rs + XPE;
  float* Pt = Qp + QPE;
  // total ws: 4*XPE + 2*QPE floats ~= 332 MB

  const int padzBlocks = (int)(XPE / 256);                         // exact
  const int padpqBlocks = (int)(QPE / 256);                        // exact
  const size_t attnLds = (size_t)(3 * 64 * LDSW + 64 * 68) * sizeof(float);  // ~236 KB < 320 KB

  pad_z_kernel<<<padzBlocks, 256, 0, stream>>>(Z, Xp);
  pad_pq_kernel<<<padpqBlocks, 256, 0, stream>>>(ap, Qp, Pt);

  for (int l = 0; l < NLAY; ++l) {
    for (int h = 0; h < NHEAD; ++h) {
      const float* q = Qp + (size_t)(l * NHEAD + h) * PADD * PADD;
      const float* p = Pt + (size_t)(l * NHEAD + h) * PADD * PADD;
      proj_kernel<<<dim3(153, BB), 256, 0, stream>>>(Xp, q, ZQ, 0);
      proj_kernel<<<dim3(153, BB), 256, 0, stream>>>(Xp, p, Kb, 1);
      attn_kernel<<<dim3(PADN / 64, BB), 256, attnLds, stream>>>(ZQ, Xp, Kb, Rs, h > 0 ? 1 : 0);
    }
    resid_kernel<<<padzBlocks, 256, 0, stream>>>(Xp, Rs);
  }

  const size_t outN = (size_t)BB * NP1 * DP1;
  unpad_kernel<<<(int)((outN + 255) / 256), 256, 0, stream>>>(Xp, out);
}